// GraphFeatureExtractor_53549652247256
// MI455X (gfx1250) — compile-verified
//
#include <hip/hip_runtime.h>

typedef float v2f  __attribute__((ext_vector_type(2)));
typedef float v4f  __attribute__((ext_vector_type(4)));
typedef float v8f  __attribute__((ext_vector_type(8)));

constexpr int B_ = 8;
constexpr int N_ = 16384;
constexpr int D_ = 128;
constexpr int H_ = 128;
constexpr int O_ = 128;
constexpr int E_ = 262144;

// ---------------------------------------------------------------------------
// K0: zero workspace (deg + msum + acc1). Harness does not re-poison between
// replays, so we must re-zero every launch.
// ---------------------------------------------------------------------------
__global__ void __launch_bounds__(256) k_zero(float* __restrict__ p, long n) {
    long i = (long)blockIdx.x * blockDim.x + threadIdx.x;
    long stride = (long)gridDim.x * blockDim.x;
    for (; i < n; i += stride) p[i] = 0.0f;
}

// ---------------------------------------------------------------------------
// K1: in-degree per (graph, node):  deg[b][dst] += 1
// ---------------------------------------------------------------------------
__global__ void __launch_bounds__(256) k_deg(const int* __restrict__ ei,
                                             float* __restrict__ deg) {
    long i = (long)blockIdx.x * blockDim.x + threadIdx.x;
    if (i >= (long)B_ * E_) return;
    int b = (int)(i >> 18);            // E_ = 2^18
    int e = (int)(i & (E_ - 1));
    int dst = ei[(long)b * 2 * E_ + E_ + e];
    atomicAdd(&deg[b * N_ + dst], 1.0f);
}

// ---------------------------------------------------------------------------
// K2: acc[b][dst][:] += x[b][src][:]   (one wave per edge, float4 per lane)
// Gather is a coalesced 512B row read; scatter is 4 f32 atomics per lane into
// the 64MB accumulator, which stays resident in the 192MB L2.
// ---------------------------------------------------------------------------
__global__ void __launch_bounds__(256) k_scatter(const float* __restrict__ x,
                                                 const int* __restrict__ ei,
                                                 float* __restrict__ acc) {
    int lane = threadIdx.x & 31;
    int wid  = threadIdx.x >> 5;
    long eg = (long)blockIdx.x * 8 + wid;      // [0, B_*E_)
    int b = (int)(eg >> 18);
    int e = (int)(eg & (E_ - 1));
    const int* eb = ei + (long)b * 2 * E_;
    int src = eb[e];
    int dst = eb[E_ + e];
    const v4f* xs = (const v4f*)(x + ((long)b * N_ + src) * D_);
    float* ad = acc + ((long)b * N_ + dst) * D_ + lane * 4;
    v4f v = xs[lane];
    atomicAdd(ad + 0, v.x);
    atomicAdd(ad + 1, v.y);
    atomicAdd(ad + 2, v.z);
    atomicAdd(ad + 3, v.w);
}

// ---------------------------------------------------------------------------
// K3: fused layer 1:  h = relu( ((acc + x) / (deg+1)) @ W1 + b1 )
// One block = one 16-row strip of one graph. 8 waves, wave w computes output
// columns [16w, 16w+16) with V_WMMA_F32_16X16X4_F32 (fp32-exact).
// h overwrites acc in place (block owns its strip exclusively).
// ---------------------------------------------------------------------------
__global__ void __launch_bounds__(256) k_layer1(const float* __restrict__ x,
                                                const float* __restrict__ deg,
                                                const float* __restrict__ W1,
                                                const float* __restrict__ b1,
                                                float* __restrict__ acc) {
    __shared__ float sA[16 * 132];   // stride 132: rows land on distinct banks
    __shared__ float sW[16];

    int t = threadIdx.x;
    int tile = blockIdx.x;                 // 0 .. B_*(N_/16)-1
    int b  = tile / (N_ / 16);
    int r0 = (tile % (N_ / 16)) * 16;

    const float* degb = deg + (long)b * N_;
    float*       accb = acc + ((long)b * N_ + r0) * D_;
    const float* xb   = x   + ((long)b * N_ + r0) * D_;

    if (t < 16) sW[t] = 1.0f / (degb[r0 + t] + 1.0f);
    __syncthreads();

    // Stage A = (acc + x) * 1/(deg+1) into LDS (coalesced: 256 thr x 8 elems)
    for (int i = 0; i < 8; ++i) {
        int idx = t + 256 * i;
        int r = idx >> 7, c = idx & 127;
        sA[r * 132 + c] = (accb[idx] + xb[idx]) * sW[r];
    }
    __syncthreads();

    int lane = t & 31, wid = t >> 5;
    int lm = lane & 15, lh = lane >> 4;
    int col0 = wid * 16;

    // C layout: vgpr j, lane l -> M = j + 8*(l/16), N = l%16.
    // Bias depends only on N, so every accumulator slot of a lane gets b1[col].
    float bias = b1[col0 + lm];
    v8f c;
    for (int j = 0; j < 8; ++j) c[j] = bias;

    const float* sArow = sA + lm * 132;
    for (int k0 = 0; k0 < D_; k0 += 4) {
        int k = k0 + 2 * lh;           // A/B frag: K = k0 + 2*(lane/16) + v
        v2f a, bb;
        a[0]  = sArow[k];
        a[1]  = sArow[k + 1];
        bb[0] = W1[k * H_ + col0 + lm];
        bb[1] = W1[(k + 1) * H_ + col0 + lm];
        c = __builtin_amdgcn_wmma_f32_16x16x4_f32(
                /*neg_a=*/false, a, /*neg_b=*/false, bb,
                /*c_mod=*/(short)0, c, /*reuse_a=*/false, /*reuse_b=*/false);
    }

    // ReLU + in-place store of h over acc strip
    for (int j = 0; j < 8; ++j) {
        int r = j + 8 * lh;
        float v = c[j];
        accb[r * D_ + col0 + lm] = v > 0.0f ? v : 0.0f;
    }
}

// ---------------------------------------------------------------------------
// K4: msum[b][:] += sum over items of h[src]*w(dst), items = E edges plus
// N virtual self-loops (src=dst=n). Register accumulation -> LDS ds_add_f32
// cross-wave reduce -> 128 global atomics per block.
// ---------------------------------------------------------------------------
__global__ void __launch_bounds__(256) k_pool(const float* __restrict__ h,
                                              const float* __restrict__ deg,
                                              const int* __restrict__ ei,
                                              float* __restrict__ msum) {
    __shared__ float red[128];
    int b = blockIdx.y;
    int lane = threadIdx.x & 31, wid = threadIdx.x >> 5;
    const int*   eb   = ei  + (long)b * 2 * E_;
    const float* hb   = h   + (long)b * N_ * H_;
    const float* degb = deg + (long)b * N_;

    v4f acc = {0.0f, 0.0f, 0.0f, 0.0f};
    const long total = (long)E_ + N_;
    for (long it = (long)blockIdx.x * 8 + wid; it < total;
         it += (long)gridDim.x * 8) {
        int src, dst;
        if (it < E_) { src = eb[it]; dst = eb[E_ + it]; }
        else         { src = (int)(it - E_); dst = src; }
        float w = 1.0f / (degb[dst] + 1.0f);
        v4f v = ((const v4f*)(hb + (long)src * H_))[lane];
        acc += v * w;
    }

    if (threadIdx.x < 128) red[threadIdx.x] = 0.0f;
    __syncthreads();
    int cb = lane * 4;
    atomicAdd(&red[cb + 0], acc.x);
    atomicAdd(&red[cb + 1], acc.y);
    atomicAdd(&red[cb + 2], acc.z);
    atomicAdd(&red[cb + 3], acc.w);
    __syncthreads();
    if (threadIdx.x < 128)
        atomicAdd(&msum[b * 128 + threadIdx.x], red[threadIdx.x]);
}

// ---------------------------------------------------------------------------
// K5: out[b] = (msum[b] / N) @ W2 + b2     (tiny: 8 x 128 x 128)
// ---------------------------------------------------------------------------
__global__ void __launch_bounds__(128) k_out(const float* __restrict__ msum,
                                             const float* __restrict__ W2,
                                             const float* __restrict__ b2,
                                             float* __restrict__ out) {
    int b = blockIdx.x;
    int o = threadIdx.x;
    const float* m = msum + b * 128;
    float s = 0.0f;
    for (int k = 0; k < H_; ++k) s += m[k] * W2[k * O_ + o];
    out[b * O_ + o] = b2[o] + s * (1.0f / (float)N_);
}

// ---------------------------------------------------------------------------
extern "C" void kernel_launch(void* const* d_in, const int* in_sizes, int n_in,
                              void* d_out, int out_size, void* d_ws, size_t ws_size,
                              hipStream_t stream) {
    const float* x  = (const float*)d_in[0];   // [B,N,D]
    const int*   ei = (const int*)d_in[1];     // [B,2E]
    const float* W1 = (const float*)d_in[4];   // [D,H]
    const float* b1 = (const float*)d_in[5];   // [H]
    const float* W2 = (const float*)d_in[6];   // [H,O]
    const float* b2 = (const float*)d_in[7];   // [O]
    float* out = (float*)d_out;                // [B,O]

    float* ws   = (float*)d_ws;
    float* deg  = ws;                          // B*N
    float* msum = deg + (long)B_ * N_;         // B*128
    float* acc  = msum + (long)B_ * 128;       // B*N*D  (acc1, then h in place)

    long zn = (long)B_ * N_ + (long)B_ * 128 + (long)B_ * N_ * D_;
    k_zero<<<4096, 256, 0, stream>>>(ws, zn);

    k_deg<<<(B_ * E_ + 255) / 256, 256, 0, stream>>>(ei, deg);

    k_scatter<<<(B_ * E_) / 8, 256, 0, stream>>>(x, ei, acc);

    k_layer1<<<B_ * (N_ / 16), 256, 0, stream>>>(x, deg, W1, b1, acc);

    dim3 gpool(2048, B_);
    k_pool<<<gpool, 256, 0, stream>>>(acc, deg, ei, msum);

    k_out<<<B_, 128, 0, stream>>>(msum, W2, b2, out);
}